// KPConv_24670292148502
// MI455X (gfx1250) — compile-verified
//
#include <hip/hip_runtime.h>

typedef float v2f __attribute__((ext_vector_type(2)));
typedef float v8f __attribute__((ext_vector_type(8)));

#define KPTS 15
#define FDIM 16
#define ODIM 16
#define INV_EXT (1.0f / 0.6f)

// Pull a wave-uniform float into an SGPR to save VGPR budget.
__device__ __forceinline__ float rfl_f32(float x) {
  return __builtin_bit_cast(
      float, __builtin_amdgcn_readfirstlane(__builtin_bit_cast(int, x)));
}

__global__ __launch_bounds__(256) void kpconv_zero(float* __restrict__ out, int n) {
  int i = blockIdx.x * blockDim.x + threadIdx.x;
  if (i < n) out[i] = 0.0f;
}

// One wave32 processes tiles of 16 edges (grid-stride over tiles).
//   A[16 x 240]: A[m, kp*16+i] = h[m][kp] * feats[src[m]][i]
//   B[240 x 16]: W reshaped;  D = A*B accumulated over 60 f32 WMMA (K=4) steps.
// B lives in 120 VGPRs for the whole kernel; the preload is amortized over
// ~12 tiles per wave (fixed 2048-block grid).
__global__ __launch_bounds__(256) void kpconv_wmma(
    const int* __restrict__ srcI, const int* __restrict__ dstI,
    const float* __restrict__ pos, const float* __restrict__ feats,
    const float* __restrict__ kpnt, const float* __restrict__ Wf,
    float* __restrict__ out, int numTiles) {
  const int lane = threadIdx.x & 31;
  const int g = lane >> 4;   // half-wave: 0 -> K cols {0,1}, 1 -> {2,3} of chunk
  const int m = lane & 15;   // edge row within tile (A) / out column (B,D)
  const int wavesPerBlk = blockDim.x >> 5;
  int tile = blockIdx.x * wavesPerBlk + (threadIdx.x >> 5);
  const int tileStride = gridDim.x * wavesPerBlk;

  // ---- Hoist B = W[240][16] into registers (layout for V_WMMA_F32_16X16X4_F32:
  //      vgpr0: rows {4kk+0 | 4kk+2}, vgpr1: rows {4kk+1 | 4kk+3}, lanes = N). ----
  v2f B[60];
#pragma unroll
  for (int kk = 0; kk < 60; ++kk) {
    const int r0 = 4 * kk + 2 * g;
    v2f b;
    b.x = Wf[r0 * ODIM + m];
    b.y = Wf[(r0 + 1) * ODIM + m];
    B[kk] = b;
  }

  // ---- Kernel points: wave-uniform -> SGPRs. ----
  float kx[KPTS], ky[KPTS], kz[KPTS];
#pragma unroll
  for (int k = 0; k < KPTS; ++k) {
    kx[k] = rfl_f32(kpnt[3 * k + 0]);
    ky[k] = rfl_f32(kpnt[3 * k + 1]);
    kz[k] = rfl_f32(kpnt[3 * k + 2]);
  }

  for (; tile < numTiles; tile += tileStride) {
    const int e = tile * 16 + m;
    const int s = srcI[e];
    const int d = dstI[e];

    // Relative position (pos is L2-resident: 1.2 MB).
    const float yx = pos[s * 3 + 0] - pos[d * 3 + 0];
    const float yy = pos[s * 3 + 1] - pos[d * 3 + 1];
    const float yz = pos[s * 3 + 2] - pos[d * 3 + 2];

    // Kernel-point influences h[m][k].
    float h[KPTS];
#pragma unroll
    for (int k = 0; k < KPTS; ++k) {
      const float dx = yx - kx[k];
      const float dy = yy - ky[k];
      const float dz = yz - kz[k];
      const float dist = sqrtf(fmaf(dx, dx, fmaf(dy, dy, dz * dz)));
      h[k] = fmaxf(0.0f, 1.0f - dist * INV_EXT);
    }

    // This half-wave only needs f[4p + 2g + {0,1}], p=0..3 -> four b64 gathers
    // (exactly 64 B/edge across the wave, no duplication).
    const float2* fp = (const float2*)(feats + (size_t)s * FDIM + 2 * g);
    const float2 q0 = fp[0], q1 = fp[2], q2 = fp[4], q3 = fp[6];
    const float fa[8] = {q0.x, q0.y, q1.x, q1.y, q2.x, q2.y, q3.x, q3.y};

    v8f acc = {};
#pragma unroll
    for (int kk = 0; kk < 60; ++kk) {
      const int kp = kk >> 2;  // compile-time: chunk never straddles a kernel pt
      const int p = kk & 3;
      v2f a;
      a.x = h[kp] * fa[2 * p + 0];
      a.y = h[kp] * fa[2 * p + 1];
      acc = __builtin_amdgcn_wmma_f32_16x16x4_f32(
          /*neg_a=*/false, a, /*neg_b=*/false, B[kk],
          /*c_mod=*/(short)0, acc, /*reuse_a=*/false, /*reuse_b=*/false);
    }

    // Scatter: acc[j] is D[M= g*8+j][N= m]. dst[M] lives in lane M -> bpermute.
    // out (6.4 MB) is L2-resident, so these f32 atomics RMW in L2.
#pragma unroll
    for (int j = 0; j < 8; ++j) {
      const int row = g * 8 + j;
      const int dj = __builtin_amdgcn_ds_bpermute(row << 2, d);
      __hip_atomic_fetch_add(out + (size_t)dj * ODIM + m, acc[j],
                             __ATOMIC_RELAXED, __HIP_MEMORY_SCOPE_AGENT);
    }
  }
}

// Scalar fallback for E % 16 tail edges (not used for E = 3.2M, kept for safety).
__global__ __launch_bounds__(256) void kpconv_tail(
    const int* __restrict__ srcI, const int* __restrict__ dstI,
    const float* __restrict__ pos, const float* __restrict__ feats,
    const float* __restrict__ kpnt, const float* __restrict__ Wf,
    float* __restrict__ out, int eStart, int E) {
  const int t = blockIdx.x * blockDim.x + threadIdx.x;
  const int e = eStart + (t >> 4);
  const int o = t & 15;
  if (e >= E) return;
  const int s = srcI[e];
  const int d = dstI[e];
  const float yx = pos[s * 3 + 0] - pos[d * 3 + 0];
  const float yy = pos[s * 3 + 1] - pos[d * 3 + 1];
  const float yz = pos[s * 3 + 2] - pos[d * 3 + 2];
  float acc = 0.0f;
  for (int k = 0; k < KPTS; ++k) {
    const float dx = yx - kpnt[3 * k + 0];
    const float dy = yy - kpnt[3 * k + 1];
    const float dz = yz - kpnt[3 * k + 2];
    const float dist = sqrtf(fmaf(dx, dx, fmaf(dy, dy, dz * dz)));
    const float h = fmaxf(0.0f, 1.0f - dist * INV_EXT);
    if (h > 0.0f) {
      float dotv = 0.0f;
      for (int i = 0; i < FDIM; ++i)
        dotv = fmaf(feats[(size_t)s * FDIM + i], Wf[(k * FDIM + i) * ODIM + o], dotv);
      acc = fmaf(h, dotv, acc);
    }
  }
  __hip_atomic_fetch_add(out + (size_t)d * ODIM + o, acc,
                         __ATOMIC_RELAXED, __HIP_MEMORY_SCOPE_AGENT);
}

extern "C" void kernel_launch(void* const* d_in, const int* in_sizes, int n_in,
                              void* d_out, int out_size, void* d_ws, size_t ws_size,
                              hipStream_t stream) {
  const int* src = (const int*)d_in[0];
  const int* dst = (const int*)d_in[1];
  const float* pos = (const float*)d_in[2];
  const float* feats = (const float*)d_in[3];
  const float* kpnt = (const float*)d_in[4];
  const float* Wf = (const float*)d_in[5];
  float* out = (float*)d_out;

  const int E = in_sizes[0];
  const int numTiles = E / 16;
  const int tail = E - numTiles * 16;

  // d_out is poisoned by the harness -> zero it before atomic accumulation.
  kpconv_zero<<<(out_size + 255) / 256, 256, 0, stream>>>(out, out_size);

  if (numTiles > 0) {
    const int wavesPerBlk = 8;  // 256 threads = 8 wave32
    // Fixed grid + grid-stride loop: ~12 tiles per wave amortizes the
    // 120-VGPR B preload and kernel-point setup; 16K waves still saturate.
    int blocks = (numTiles + wavesPerBlk - 1) / wavesPerBlk;
    if (blocks > 2048) blocks = 2048;
    kpconv_wmma<<<blocks, 256, 0, stream>>>(src, dst, pos, feats, kpnt, Wf, out,
                                            numTiles);
  }
  if (tail > 0) {
    kpconv_tail<<<(tail * 16 + 255) / 256, 256, 0, stream>>>(
        src, dst, pos, feats, kpnt, Wf, out, numTiles * 16, E);
  }
}